// GraphReinforceAgent_29368986370400
// MI455X (gfx1250) — compile-verified
//
#include <hip/hip_runtime.h>
#include <math.h>

// ---------------------------------------------------------------------------
// GCNConv(2->256) + ReLU + LayerNorm + global_add_pool + MLP head, MI455X.
//
// Key algebraic optimization: agg = A_hat @ (X @ W) == (A_hat @ X) @ W.
// Aggregate the 2-wide raw features over edges (cheap), then do the 2->256
// projection with V_WMMA_F32_16X16X4_F32 (16 nodes x 16 hidden per WMMA,
// K=4 with features zero-padded from 2). LN + pool fused in-wave.
// ---------------------------------------------------------------------------

typedef __attribute__((ext_vector_type(2))) float v2f;
typedef __attribute__((ext_vector_type(8))) float v8f;

#define HIDDEN 256
#define LN_EPS 1e-5f
#define NODES_PER_BLOCK 128   // 8 waves x 16 nodes

// ---- kernel 0: zero deg[0..N) and pooled[0..256) ---------------------------
__global__ void k_zero(float* deg, float* pooled, int N) {
    int i = blockIdx.x * blockDim.x + threadIdx.x;
    if (i < N) deg[i] = 0.0f;
    if (i < HIDDEN) pooled[i] = 0.0f;
}

// ---- kernel 1: degree counting over edges (dst side) -----------------------
__global__ void k_degree(const int* __restrict__ ei, float* __restrict__ deg,
                         int E, int N) {
    int e = blockIdx.x * blockDim.x + threadIdx.x;
    if (e >= E) return;
    int d = ei[E + e];          // edge_index[1][e]
    if (d >= 0 && d < N) atomicAdd(&deg[d], 1.0f);
}

// ---- kernel 2: dinv = rsqrt(deg+1) (self-loop), seed agg2 with self-loop ---
__global__ void k_node_prep(const float* __restrict__ x,   // [N,2]
                            float* __restrict__ dinv,      // in: deg, out: dinv
                            float* __restrict__ agg2,      // [N,2]
                            int N) {
    int i = blockIdx.x * blockDim.x + threadIdx.x;
    if (i >= N) return;
    float di = rsqrtf(dinv[i] + 1.0f);   // +1: self loop; always > 0
    dinv[i] = di;
    float nrm = di * di;                 // self-loop norm = dinv[i]^2
    agg2[2 * i + 0] = nrm * x[2 * i + 0];
    agg2[2 * i + 1] = nrm * x[2 * i + 1];
}

// ---- kernel 3: per-edge 2-float aggregation --------------------------------
__global__ void k_edge_agg(const int* __restrict__ ei,
                           const float* __restrict__ x,
                           const float* __restrict__ dinv,
                           float* __restrict__ agg2,
                           int E, int N) {
    int e = blockIdx.x * blockDim.x + threadIdx.x;
    if (e >= E) return;
    int s = ei[e];
    int d = ei[E + e];
    if (s < 0 || s >= N || d < 0 || d >= N) return;
    float nrm = dinv[s] * dinv[d];
    atomicAdd(&agg2[2 * d + 0], nrm * x[2 * s + 0]);
    atomicAdd(&agg2[2 * d + 1], nrm * x[2 * s + 1]);
}

// ---- kernel 4: WMMA projection + bias + ReLU + LayerNorm + pooled sum ------
// One wave handles 16 nodes. A tile (16x4 f32): lanes 0-15 hold K=0,1
// (the 2 real features), lanes 16-31 hold K=2,3 = 0 (pad).
// B tile (4x16 f32): lanes 0-15 hold rows K=0,1 of W_gcn, lanes 16-31 zero.
// D layout (per ISA): col = lane%16, row = vgpr + 8*(lane/16).
__global__ void __launch_bounds__(256)
k_gcn_ln_pool(const float* __restrict__ agg2,
              const float* __restrict__ Wg,      // [2,256] row-major
              const float* __restrict__ bg,      // [256]
              const float* __restrict__ gamma,
              const float* __restrict__ beta,
              float* __restrict__ pooled_g,      // [256]
              int N) {
    __shared__ float pooled_s[HIDDEN];
    pooled_s[threadIdx.x] = 0.0f;
    __syncthreads();

    const int lane = threadIdx.x & 31;
    const int wave = threadIdx.x >> 5;
    const int baseNode = blockIdx.x * NODES_PER_BLOCK + wave * 16;
    const int colLocal = lane & 15;

    if (baseNode < N) {                 // wave-uniform branch: EXEC stays full
        // ---- A tile ----
        v2f a = {0.0f, 0.0f};
        {
            int node = baseNode + colLocal;     // M = lane for lanes 0-15
            if (lane < 16 && node < N) {
                a.x = agg2[2 * node + 0];       // K=0
                a.y = agg2[2 * node + 1];       // K=1
            }
        }
        // ---- preload all 16 B tiles (W_gcn is 2x256, 2 regs/lane/tile) ----
        v2f btile[16];
        #pragma unroll
        for (int t = 0; t < 16; ++t) {
            v2f b = {0.0f, 0.0f};
            int col = t * 16 + colLocal;
            if (lane < 16) {                    // rows K=0,1 live; K=2,3 pad
                b.x = Wg[col];                  // W_gcn[0][col]
                b.y = Wg[HIDDEN + col];         // W_gcn[1][col]
            }
            btile[t] = b;
        }

        // ---- pass 1: WMMA, accumulate row stats ----
        float rs[8], rss[8];
        #pragma unroll
        for (int r = 0; r < 8; ++r) { rs[r] = 0.0f; rss[r] = 0.0f; }

        for (int t = 0; t < 16; ++t) {
            v8f c = {};
            v8f d = __builtin_amdgcn_wmma_f32_16x16x4_f32(
                        false, a, false, btile[t], (short)0, c, false, false);
            float bv = bg[t * 16 + colLocal];
            #pragma unroll
            for (int r = 0; r < 8; ++r) {
                float h = d[r] + bv;
                h = fmaxf(h, 0.0f);
                rs[r]  += h;
                rss[r] += h * h;
            }
        }
        // reduce over the 16 lanes of each half-group (rows stay in-group)
        #pragma unroll
        for (int off = 1; off < 16; off <<= 1) {
            #pragma unroll
            for (int r = 0; r < 8; ++r) {
                rs[r]  += __shfl_xor(rs[r],  off, 32);
                rss[r] += __shfl_xor(rss[r], off, 32);
            }
        }
        float mean[8], rstd[8];
        #pragma unroll
        for (int r = 0; r < 8; ++r) {
            float m = rs[r] * (1.0f / HIDDEN);
            float v = rss[r] * (1.0f / HIDDEN) - m * m;
            mean[r] = m;
            rstd[r] = rsqrtf(v + LN_EPS);
        }

        const int rowBase = baseNode + ((lane >> 4) << 3);  // +8 for hi half

        // ---- pass 2: recompute WMMA (cheap), normalize, pool ----
        for (int t = 0; t < 16; ++t) {
            v8f c = {};
            v8f d = __builtin_amdgcn_wmma_f32_16x16x4_f32(
                        false, a, false, btile[t], (short)0, c, false, false);
            int   col = t * 16 + colLocal;
            float bv = bg[col];
            float g  = gamma[col];
            float be = beta[col];
            float p = 0.0f;
            #pragma unroll
            for (int r = 0; r < 8; ++r) {
                float h = fmaxf(d[r] + bv, 0.0f);
                float nv = (h - mean[r]) * rstd[r] * g + be;
                if (rowBase + r < N) p += nv;   // mask padded rows
            }
            atomicAdd(&pooled_s[col], p);       // ds_add_f32
        }
    }
    __syncthreads();
    atomicAdd(&pooled_g[threadIdx.x], pooled_s[threadIdx.x]);
}

// ---- kernel 5: MLP head + log_softmax (single block, trivial size) ---------
__global__ void k_head(const float* __restrict__ pooled,
                       const float* __restrict__ Wh,   // [256,256]
                       const float* __restrict__ bh,
                       const float* __restrict__ Wo,   // [256,10]
                       const float* __restrict__ bo,
                       float* __restrict__ out) {
    __shared__ float z[HIDDEN];
    __shared__ float lg[16];
    __shared__ float lse_s;
    int j = threadIdx.x;
    float acc = bh[j];
    for (int k = 0; k < HIDDEN; ++k)
        acc = fmaf(pooled[k], Wh[k * HIDDEN + j], acc);
    z[j] = fmaxf(acc, 0.0f);
    __syncthreads();
    if (j < 10) {
        float a = bo[j];
        for (int k = 0; k < HIDDEN; ++k)
            a = fmaf(z[k], Wo[k * 10 + j], a);
        lg[j] = a;
    }
    __syncthreads();
    if (j == 0) {
        float m = -1e30f;
        for (int o = 0; o < 10; ++o) m = fmaxf(m, lg[o]);
        float s = 0.0f;
        for (int o = 0; o < 10; ++o) s += expf(lg[o] - m);
        lse_s = m + logf(s);
    }
    __syncthreads();
    if (j < 10) out[j] = lg[j] - lse_s;
}

// ---------------------------------------------------------------------------
extern "C" void kernel_launch(void* const* d_in, const int* in_sizes, int n_in,
                              void* d_out, int out_size, void* d_ws, size_t ws_size,
                              hipStream_t stream) {
    const float* x     = (const float*)d_in[0];   // [N,2]
    const int*   ei    = (const int*)  d_in[1];   // [2,E]
    const float* Wg    = (const float*)d_in[2];   // [2,256]
    const float* bg    = (const float*)d_in[3];
    const float* gamma = (const float*)d_in[4];
    const float* beta  = (const float*)d_in[5];
    const float* Wh    = (const float*)d_in[6];   // [256,256]
    const float* bh    = (const float*)d_in[7];
    const float* Wo    = (const float*)d_in[8];   // [256,10]
    const float* bo    = (const float*)d_in[9];
    float* out = (float*)d_out;

    const int N = in_sizes[0] / 2;
    const int E = in_sizes[1] / 2;

    // workspace layout (floats): deg/dinv[N] | agg2[2N] | pooled[256]
    float* ws     = (float*)d_ws;
    float* dinv   = ws;
    float* agg2   = ws + N;
    float* pooled = ws + 3 * (size_t)N;

    const int T = 256;
    k_zero<<<(N + T - 1) / T, T, 0, stream>>>(dinv, pooled, N);
    k_degree<<<(E + T - 1) / T, T, 0, stream>>>(ei, dinv, E, N);
    k_node_prep<<<(N + T - 1) / T, T, 0, stream>>>(x, dinv, agg2, N);
    k_edge_agg<<<(E + T - 1) / T, T, 0, stream>>>(ei, x, dinv, agg2, E, N);
    k_gcn_ln_pool<<<(N + NODES_PER_BLOCK - 1) / NODES_PER_BLOCK, T, 0, stream>>>(
        agg2, Wg, bg, gamma, beta, pooled, N);
    k_head<<<1, T, 0, stream>>>(pooled, Wh, bh, Wo, bo, out);
}